// GCN_71975061946794
// MI455X (gfx1250) — compile-verified
//
#include <hip/hip_runtime.h>
#include <hip/hip_bf16.h>

typedef float v2f __attribute__((ext_vector_type(2)));
typedef float v4f __attribute__((ext_vector_type(4)));
typedef float v8f __attribute__((ext_vector_type(8)));

#define N_NODES 100000
#define N_EDGES 1600000
#define DDIM    64
#define FPAD    68    // feature-tile LDS row stride (floats): 272B = 16B-aligned, bank-conflict-free
#define PSTRIDE 160   // K-pair LDS stride (floats): +160 ≡ +32 banks -> kh=0/kh=1 lanes use disjoint banks

// ---------------------------------------------------------------------------
// Kernel 1: h = features @ kernel  via fp32 WMMA 16x16x4
// Block = 256 threads (8 waves). One wave per 16-row node tile; each wave
// produces the full 16x64 output block (4 column tiles of 16).
//
// Weight matrix staged in LDS in K-pair-interleaved layout:
//   Kpack[(k/2)*PSTRIDE + col*2 + (k&1)] = W[k][col]
// so a lane's B fragment {W[K][col], W[K+1][col]} is ONE aligned ds_load_b64
// (kills the v_mov fragment-assembly churn seen in round 1 asm).
// ---------------------------------------------------------------------------
__global__ __launch_bounds__(256) void gcn_gemm_wmma(
    const float* __restrict__ features, const float* __restrict__ kern,
    float* __restrict__ h)
{
    __shared__ float Kpack[(DDIM / 2) * PSTRIDE];   // 32 pairs * 160 floats = 20 KB
    __shared__ float Flds[8][16 * FPAD];            // 8 waves * 4.25 KB feature tiles

    const int t = threadIdx.x;

    // cooperative load+repack of the 64x64 weight matrix (coalesced b128 reads)
    {
        const v4f* src = (const v4f*)kern;
        #pragma unroll
        for (int j = 0; j < (DDIM * DDIM / 4) / 256; ++j) {   // 4 iterations
            int i = t + j * 256;
            v4f w = src[i];
            int k  = i >> 4;            // 16 float4 per row
            int c0 = (i & 15) * 4;
            float* dst = &Kpack[(k >> 1) * PSTRIDE + (k & 1)];
            dst[(c0 + 0) * 2] = w.x;
            dst[(c0 + 1) * 2] = w.y;
            dst[(c0 + 2) * 2] = w.z;
            dst[(c0 + 3) * 2] = w.w;
        }
    }

    const int wave = t >> 5;
    const int lane = t & 31;
    const int m    = lane & 15;   // row (A) / col (B) within 16
    const int kh   = lane >> 4;   // k-half selector

    const int tile   = blockIdx.x * 8 + wave;
    const bool active = (tile * 16) < N_NODES;

    if (active) {
        // stage this wave's 16x64 feature tile (4KB contiguous) into padded LDS
        const int row0 = tile * 16;
        const v4f* src = (const v4f*)(features + (size_t)row0 * DDIM);
        #pragma unroll
        for (int j = 0; j < 8; ++j) {             // 256 float4 / 32 lanes
            int i = lane + j * 32;
            int r = i >> 4;                       // 16 float4 per row
            int c = i & 15;
            *(v4f*)&Flds[wave][r * FPAD + c * 4] = src[i];
        }
    }
    __syncthreads();
    if (!active) return;                          // wave-uniform: EXEC stays all-1s

    v8f acc0 = {}, acc1 = {}, acc2 = {}, acc3 = {};
    const float* Fw = &Flds[wave][0];

    #pragma unroll
    for (int k = 0; k < DDIM; k += 4) {
        // A fragment: VGPR0 = K=k+2kh, VGPR1 = K=k+2kh+1  (single b64 load)
        v2f a = *(const v2f*)&Fw[m * FPAD + k + 2 * kh];

        // B fragments: pair p = k/2 + kh holds rows (k+2kh, k+2kh+1)
        const float* Kp = &Kpack[((k >> 1) + kh) * PSTRIDE];
        v2f b0 = *(const v2f*)&Kp[( 0 + m) * 2];
        v2f b1 = *(const v2f*)&Kp[(16 + m) * 2];
        v2f b2 = *(const v2f*)&Kp[(32 + m) * 2];
        v2f b3 = *(const v2f*)&Kp[(48 + m) * 2];

        acc0 = __builtin_amdgcn_wmma_f32_16x16x4_f32(false, a, false, b0, (short)0, acc0, false, false);
        acc1 = __builtin_amdgcn_wmma_f32_16x16x4_f32(false, a, false, b1, (short)0, acc1, false, false);
        acc2 = __builtin_amdgcn_wmma_f32_16x16x4_f32(false, a, false, b2, (short)0, acc2, false, false);
        acc3 = __builtin_amdgcn_wmma_f32_16x16x4_f32(false, a, false, b3, (short)0, acc3, false, false);
    }

    // D layout: VGPR r, lanes 0-15 -> row r, lanes 16-31 -> row r+8; col = m
    const int row0 = tile * 16;
    #pragma unroll
    for (int r = 0; r < 8; ++r) {
        float* hp = h + (size_t)(row0 + r + 8 * kh) * DDIM;
        hp[ 0 + m] = acc0[r];
        hp[16 + m] = acc1[r];
        hp[32 + m] = acc2[r];
        hp[48 + m] = acc3[r];
    }
}

// ---------------------------------------------------------------------------
// Kernel 2: edge scatter. One wave per edge; each lane handles 2 dims.
// agg[row] += val * h[col]   (global_atomic_add_f32 resolved in L2; h is
// 25.6 MB and resident in the 192 MB L2)
// ---------------------------------------------------------------------------
__global__ __launch_bounds__(256) void gcn_edge_scatter(
    const int* __restrict__ erow, const int* __restrict__ ecol,
    const float* __restrict__ eval_, const float* __restrict__ h,
    float* __restrict__ agg)
{
    const int gtid = blockIdx.x * 256 + threadIdx.x;
    const int e    = gtid >> 5;
    const int lane = gtid & 31;
    if (e >= N_EDGES) return;

    const int   r = erow[e];
    const int   c = ecol[e];
    const float v = eval_[e];

    v2f hv = *(const v2f*)(h + (size_t)c * DDIM + lane * 2);
    float* dst = agg + (size_t)r * DDIM + lane * 2;
    atomicAdd(dst + 0, hv.x * v);
    atomicAdd(dst + 1, hv.y * v);
}

// ---------------------------------------------------------------------------
// Kernel 3: out = selu(h * skip + agg + bias), float2-vectorized, in-place on
// the agg buffer (each thread touches only its own elements).
// ---------------------------------------------------------------------------
__global__ __launch_bounds__(256) void gcn_finalize(
    const float* __restrict__ h, const float* __restrict__ bias,
    const float* __restrict__ skip, float* __restrict__ out)
{
    const float alpha = 1.6732632423543772f;
    const float scale = 1.0507009873554805f;

    const int i = blockIdx.x * 256 + threadIdx.x;   // one float2 per thread
    if (i >= N_NODES * DDIM / 2) return;
    const int dpair = i & (DDIM / 2 - 1);           // dim pair index
    const int d0 = dpair * 2;

    v2f hv = *(const v2f*)(h + (size_t)i * 2);
    v2f av = *(const v2f*)(out + (size_t)i * 2);

    float x0 = hv.x * skip[d0 + 0] + av.x + bias[d0 + 0];
    float x1 = hv.y * skip[d0 + 1] + av.y + bias[d0 + 1];

    v2f res;
    res.x = x0 > 0.0f ? scale * x0 : scale * alpha * expm1f(x0);
    res.y = x1 > 0.0f ? scale * x1 : scale * alpha * expm1f(x1);
    *(v2f*)(out + (size_t)i * 2) = res;
}

// ---------------------------------------------------------------------------
extern "C" void kernel_launch(void* const* d_in, const int* in_sizes, int n_in,
                              void* d_out, int out_size, void* d_ws, size_t ws_size,
                              hipStream_t stream) {
    const float* features = (const float*)d_in[0];   // [100000, 64]
    const float* kern     = (const float*)d_in[1];   // [64, 64]
    const float* bias     = (const float*)d_in[2];   // [64]
    const float* skip     = (const float*)d_in[3];   // [64]
    const int*   erow     = (const int*)  d_in[4];   // [1.6M]
    const int*   ecol     = (const int*)  d_in[5];   // [1.6M]
    const float* eval_    = (const float*)d_in[6];   // [1.6M]
    float* out = (float*)d_out;
    float* h   = (float*)d_ws;                       // 25.6 MB scratch for h

    // agg accumulates directly into d_out; zero it first (capture-safe)
    hipMemsetAsync(d_out, 0, (size_t)N_NODES * DDIM * sizeof(float), stream);

    const int n_tiles = N_NODES / 16;                // 6250 exactly
    gcn_gemm_wmma<<<(n_tiles + 7) / 8, 256, 0, stream>>>(features, kern, h);

    gcn_edge_scatter<<<(N_EDGES * 32) / 256, 256, 0, stream>>>(erow, ecol, eval_, h, out);

    gcn_finalize<<<(N_NODES * DDIM / 2) / 256, 256, 0, stream>>>(h, bias, skip, out);
}